// GCN_51058571215473
// MI455X (gfx1250) — compile-verified
//
#include <hip/hip_runtime.h>
#include <math.h>

typedef __attribute__((ext_vector_type(2))) float v2f;
typedef __attribute__((ext_vector_type(8))) float v8f;

#define THREADS 256

static inline size_t alignUp(size_t x, size_t a) { return (x + a - 1) & ~(a - 1); }

// ---------------------------------------------------------------------------
// degree / norm kernels
// ---------------------------------------------------------------------------
__global__ void fill_zero_kernel(float* __restrict__ p, long long n) {
    long long i = blockIdx.x * (long long)blockDim.x + threadIdx.x;
    if (i < n) p[i] = 0.0f;
}

__global__ void degree_kernel(const long long* __restrict__ dst,
                              float* __restrict__ deg, long long E) {
    long long e = blockIdx.x * (long long)blockDim.x + threadIdx.x;
    if (e < E) {
        unsafeAtomicAdd(&deg[dst[e]], 1.0f);
    }
}

__global__ void rsqrt_kernel(float* __restrict__ p, long long n) {
    long long i = blockIdx.x * (long long)blockDim.x + threadIdx.x;
    if (i < n) {
        float d = p[i] + 1.0f;          // + self loop
        p[i] = 1.0f / sqrtf(d);         // deg^{-1/2}, deg >= 1 always
    }
}

// ---------------------------------------------------------------------------
// Dense transform H = relu?(X) @ W via V_WMMA_F32_16X16X4_F32.
// One wave computes a 16-row slab and all (COUT+15)/16 column tiles.
// A frag (16x4 f32): lane L -> row = L&15, K = k0 + 2*(L>>4) + {0,1} in v2f.
// B frag (4x16 f32): lane L -> col = L&15, same K pair.
// C/D (16x16 f32):   VGPR v -> M = v + 8*(L>>4), N = L&15.
// All loads are branchless (clamped indices + value selects) so EXEC stays
// full through the WMMA chain and no exec-save/restore code is generated.
// ---------------------------------------------------------------------------
template <int CIN, int COUT, bool RELU_IN>
__global__ void gemm_wmma_kernel(const float* __restrict__ X,
                                 const float* __restrict__ W,
                                 float* __restrict__ H, long long N) {
    const int lane = threadIdx.x & 31;
    const int wave = threadIdx.x >> 5;                             // 0..7
    const long long row0 = ((long long)blockIdx.x * 8 + wave) * 16;
    if (row0 >= N) return;                                         // wave-uniform

    const int n0 = lane & 15;                                      // col within tile
    const int kb = (lane >> 4) * 2;                                // 0 or 2
    long long row = row0 + n0;                                     // A row = lane&15
    if (row >= N) row = N - 1;                                     // clamp (stores guarded)
    const float* __restrict__ Xr = X + row * CIN;

    constexpr int KI = (CIN + 3) / 4;                              // k-steps of 4
    constexpr int NT = (COUT + 15) / 16;                           // 16-col tiles

    // ---- hoist all B fragments (lane-invariant across rows, tiny) ----
    v2f bf[KI][NT];
#pragma unroll
    for (int ki = 0; ki < KI; ++ki) {
        const int ka = ki * 4 + kb;
#pragma unroll
        for (int t = 0; t < NT; ++t) {
            const int n = t * 16 + n0;
            const bool ok0 = (ka < CIN) && (n < COUT);
            const bool ok1 = (ka + 1 < CIN) && (n < COUT);
            const float w0 = W[ok0 ? (ka * COUT + n) : 0];
            const float w1 = W[ok1 ? ((ka + 1) * COUT + n) : 0];
            bf[ki][t].x = ok0 ? w0 : 0.0f;
            bf[ki][t].y = ok1 ? w1 : 0.0f;
        }
    }

    v8f acc[NT] = {};
#pragma unroll
    for (int ki = 0; ki < KI; ++ki) {
        const int ka = ki * 4 + kb;
        float a0 = Xr[(ka < CIN) ? ka : 0];
        float a1 = Xr[(ka + 1 < CIN) ? (ka + 1) : 0];
        if (RELU_IN) { a0 = fmaxf(a0, 0.0f); a1 = fmaxf(a1, 0.0f); }
        v2f a;
        a.x = (ka < CIN) ? a0 : 0.0f;
        a.y = (ka + 1 < CIN) ? a1 : 0.0f;
#pragma unroll
        for (int t = 0; t < NT; ++t) {
            acc[t] = __builtin_amdgcn_wmma_f32_16x16x4_f32(
                false, a, false, bf[ki][t], (short)0, acc[t], false, false);
        }
    }

    // ---- store D tiles ----
    const int mo = (lane >> 4) * 8;
#pragma unroll
    for (int t = 0; t < NT; ++t) {
        const int n = t * 16 + n0;
        if (n >= COUT) continue;
#pragma unroll
        for (int v = 0; v < 8; ++v) {
            const long long r = row0 + mo + v;
            if (r < N) H[r * COUT + n] = acc[t][v];
        }
    }
}

// ---------------------------------------------------------------------------
// agg = h * dis^2 (self loop) + bias   (fully initializes agg buffer)
// ---------------------------------------------------------------------------
__global__ void init_agg_kernel(const float* __restrict__ h,
                                const float* __restrict__ dis,
                                const float* __restrict__ bias,
                                float* __restrict__ agg,
                                long long N, int cShift) {
    long long idx = blockIdx.x * (long long)blockDim.x + threadIdx.x;
    if (idx >= (N << cShift)) return;
    long long i = idx >> cShift;
    int c = (int)(idx & ((1LL << cShift) - 1));
    float d = dis[i];
    agg[idx] = h[idx] * d * d + bias[c];
}

// ---------------------------------------------------------------------------
// per (edge, channel): agg[dst,c] += h[src,c] * dis[src]*dis[dst]
// wave32 covers 32 channels of one edge at C=32 -> broadcast idx/norm loads,
// coalesced 128B gather, hardware global_atomic_add_f32 into L2-resident agg.
// ---------------------------------------------------------------------------
__global__ void edge_scatter_kernel(const long long* __restrict__ src,
                                    const long long* __restrict__ dst,
                                    const float* __restrict__ dis,
                                    const float* __restrict__ h,
                                    float* __restrict__ agg,
                                    long long E, int cShift) {
    long long idx = blockIdx.x * (long long)blockDim.x + threadIdx.x;
    if (idx >= (E << cShift)) return;
    long long e = idx >> cShift;
    int c = (int)(idx & ((1LL << cShift) - 1));
    long long s = src[e];
    long long d = dst[e];
    float w = dis[s] * dis[d];
    unsafeAtomicAdd(&agg[(d << cShift) + c], h[(s << cShift) + c] * w);
}

// ---------------------------------------------------------------------------
// 2-class log_softmax per node
// ---------------------------------------------------------------------------
__global__ void log_softmax2_kernel(const float* __restrict__ agg,
                                    float* __restrict__ out, long long N) {
    long long i = blockIdx.x * (long long)blockDim.x + threadIdx.x;
    if (i >= N) return;
    float a0 = agg[2 * i];
    float a1 = agg[2 * i + 1];
    float m  = fmaxf(a0, a1);
    float lse = m + logf(expf(a0 - m) + expf(a1 - m));
    out[2 * i]     = a0 - lse;
    out[2 * i + 1] = a1 - lse;
}

// ---------------------------------------------------------------------------
static inline int nblocks(long long total, int t) {
    return (int)((total + t - 1) / t);
}

extern "C" void kernel_launch(void* const* d_in, const int* in_sizes, int n_in,
                              void* d_out, int out_size, void* d_ws, size_t ws_size,
                              hipStream_t stream) {
    const float*     x  = (const float*)d_in[0];
    const long long* ei = (const long long*)d_in[1];
    const float*     W1 = (const float*)d_in[2];
    const float*     b1 = (const float*)d_in[3];
    const float*     W2 = (const float*)d_in[4];
    const float*     b2 = (const float*)d_in[5];
    const float*     W3 = (const float*)d_in[6];
    const float*     b3 = (const float*)d_in[7];
    float* out = (float*)d_out;

    const long long N = in_sizes[0] / 55;
    const long long E = in_sizes[1] / 2;
    const long long* src = ei;       // edge_index[0]
    const long long* dst = ei + E;   // edge_index[1]

    char* ws = (char*)d_ws;
    size_t off = 0;
    float* dis    = (float*)(ws + off); off += alignUp((size_t)N * 4, 256);
    float* bufH   = (float*)(ws + off); off += alignUp((size_t)N * 32 * 4, 256);
    float* bufAgg = (float*)(ws + off);
    (void)ws_size; (void)n_in; (void)out_size;

    const int slabBlocks = (int)(((N + 15) / 16 + 7) / 8); // 8 waves/block, 16 rows/wave

    // ---- normalization: deg -> deg^{-1/2} ----
    fill_zero_kernel<<<nblocks(N, THREADS), THREADS, 0, stream>>>(dis, N);
    degree_kernel<<<nblocks(E, THREADS), THREADS, 0, stream>>>(dst, dis, E);
    rsqrt_kernel<<<nblocks(N, THREADS), THREADS, 0, stream>>>(dis, N);

    // ---- layer 1: 55 -> 32 (ReLU fused into layer-2 GEMM load) ----
    gemm_wmma_kernel<55, 32, false><<<slabBlocks, THREADS, 0, stream>>>(x, W1, bufH, N);
    init_agg_kernel<<<nblocks(N << 5, THREADS), THREADS, 0, stream>>>(bufH, dis, b1, bufAgg, N, 5);
    edge_scatter_kernel<<<nblocks(E << 5, THREADS), THREADS, 0, stream>>>(src, dst, dis, bufH, bufAgg, E, 5);

    // ---- layer 2: 32 -> 16 (ReLU on input, fused) ----
    gemm_wmma_kernel<32, 16, true><<<slabBlocks, THREADS, 0, stream>>>(bufAgg, W2, bufH, N);
    init_agg_kernel<<<nblocks(N << 4, THREADS), THREADS, 0, stream>>>(bufH, dis, b2, bufAgg, N, 4);
    edge_scatter_kernel<<<nblocks(E << 4, THREADS), THREADS, 0, stream>>>(src, dst, dis, bufH, bufAgg, E, 4);

    // ---- layer 3: 16 -> 2 (ReLU on input, fused), then log_softmax ----
    gemm_wmma_kernel<16, 2, true><<<slabBlocks, THREADS, 0, stream>>>(bufAgg, W3, bufH, N);
    init_agg_kernel<<<nblocks(N << 1, THREADS), THREADS, 0, stream>>>(bufH, dis, b3, bufAgg, N, 1);
    edge_scatter_kernel<<<nblocks(E << 1, THREADS), THREADS, 0, stream>>>(src, dst, dis, bufH, bufAgg, E, 1);
    log_softmax2_kernel<<<nblocks(N, THREADS), THREADS, 0, stream>>>(bufAgg, out, N);
}